// LFR_21217138442633
// MI455X (gfx1250) — compile-verified
//
#include <hip/hip_runtime.h>
#include <hip/hip_bf16.h>
#include <math.h>

typedef float v2f __attribute__((ext_vector_type(2)));
typedef float v8f __attribute__((ext_vector_type(8)));

#define NN      65536
#define DD      512
#define KK      5
#define WAVES   8            // 256 threads
#define ROWS_PB (WAVES * 16) // 128 rows per block

// out layout (concatenated flat, reference return order):
//   mapping (N*K) | reconstructed (N*D) | pred (N)

__global__ __launch_bounds__(256) void lfr_fused_kernel(
    const float* __restrict__ x,        // (N, D)
    const float* __restrict__ alpha,    // (D,)
    const float* __restrict__ clw,      // (K,1) flat
    const float* __restrict__ cent,     // (K, D)
    float* __restrict__ out_map,        // (N, K)
    float* __restrict__ out_rec,        // (N, D)
    float* __restrict__ out_pred)       // (N,)
{
    // LDS: Bp = alpha-scaled centroids, zero-padded to 16 cols, layout [d][16]
    __shared__ __align__(16) float s_bp[DD * 16];      // 32 KB
    __shared__ __align__(16) float s_cent[KK * DD];    // 10 KB
    __shared__ float s_r[KK];                          // r[k] = sum_d alpha*c^2
    __shared__ float s_sw[KK];                         // sigmoid(classif_w)
    __shared__ float s_g[WAVES][16][16];               // 8 KB  (G tiles)
    __shared__ float s_m[WAVES][16][KK];               // 2.5 KB (softmax weights)

    const int tid  = threadIdx.x;
    const int wave = tid >> 5;
    const int lane = tid & 31;
    const int n16  = lane & 15;     // N index within tile / row index for A
    const int hi   = lane >> 4;     // half-wave selects K=0,1 vs K=2,3
    const int rowbase = blockIdx.x * ROWS_PB + wave * 16;

    // ---------- Phase 0: build B, copy centroids, per-k constants ----------
    for (int idx = tid; idx < DD * 16; idx += 256) {
        const int d = idx >> 4, n = idx & 15;
        s_bp[idx] = (n < KK) ? alpha[d] * cent[n * DD + d] : 0.0f;
    }
    for (int idx = tid; idx < KK * DD; idx += 256)
        s_cent[idx] = cent[idx];
    if (tid < KK) {
        float acc = 0.0f;
        for (int d = 0; d < DD; ++d) {
            const float c = cent[tid * DD + d];
            acc = fmaf(alpha[d] * c, c, acc);
        }
        s_r[tid]  = acc;
        s_sw[tid] = 1.0f / (1.0f + __expf(-clw[tid]));
    }
    __syncthreads();

    // ---------- Phase 1: G(16x16) = X_tile(16x512) @ Bp^T via WMMA f32 ----------
    // A layout (16x4 f32): lanes 0-15 hold (M=lane, K=0/1), lanes 16-31 (M=lane-16, K=2/3)
    // B layout (4x16 f32): VGPR0 = rows K=0 (lanes 0-15) / K=2 (16-31), VGPR1 = K=1 / K=3
    v8f acc = {};
    const float* aptr = x + (size_t)(rowbase + n16) * DD + (hi * 2);
#pragma unroll 4
    for (int i = 0; i < DD / 4; ++i) {
        const float2 av = *(const float2*)(aptr + 4 * i);
        v2f a; a.x = av.x; a.y = av.y;
        const int db = i * 4 + hi * 2;
        v2f b; b.x = s_bp[db * 16 + n16];
               b.y = s_bp[(db + 1) * 16 + n16];
        acc = __builtin_amdgcn_wmma_f32_16x16x4_f32(
            /*neg_a=*/false, a, /*neg_b=*/false, b,
            /*c_mod=*/(short)0, acc, /*reuse_a=*/false, /*reuse_b=*/false);
    }

    // Spill D tile: VGPR i -> (M = i + 8*hi, N = n16)
#pragma unroll
    for (int i = 0; i < 8; ++i)
        s_g[wave][i + 8 * hi][n16] = acc[i];
    __syncthreads();

    // ---------- Phase 2: softmax over K=5 (shift-invariant: q[n] dropped) ----------
    if (lane < 16) {
        const int row = rowbase + lane;
        float s[KK];
        float mx = -INFINITY;
#pragma unroll
        for (int k = 0; k < KK; ++k) {
            s[k] = fmaf(-2.0f, s_g[wave][lane][k], s_r[k]);
            mx = fmaxf(mx, s[k]);
        }
        float sum = 0.0f;
#pragma unroll
        for (int k = 0; k < KK; ++k) { s[k] = expf(s[k] - mx); sum += s[k]; }
        const float inv = 1.0f / sum;
        float pred = 0.0f;
#pragma unroll
        for (int k = 0; k < KK; ++k) {
            const float m = s[k] * inv;
            out_map[(size_t)row * KK + k] = m;
            s_m[wave][lane][k] = m;
            pred = fmaf(m, s_sw[k], pred);
        }
        out_pred[row] = pred;
    }
    __syncthreads();

    // ---------- Phase 3: reconstructed = mapping @ centroids (coalesced f4 stores) ----------
    float4* rec4 = (float4*)out_rec;
    const float4* c4 = (const float4*)s_cent;
#pragma unroll
    for (int it = 0; it < 4; ++it) {
        const int d4 = it * 32 + lane;            // float4 column index, 0..127
        float4 c[KK];
#pragma unroll
        for (int k = 0; k < KK; ++k) c[k] = c4[k * (DD / 4) + d4];
        for (int rl = 0; rl < 16; ++rl) {
            float4 o = make_float4(0.f, 0.f, 0.f, 0.f);
#pragma unroll
            for (int k = 0; k < KK; ++k) {
                const float m = s_m[wave][rl][k];
                o.x = fmaf(m, c[k].x, o.x);
                o.y = fmaf(m, c[k].y, o.y);
                o.z = fmaf(m, c[k].z, o.z);
                o.w = fmaf(m, c[k].w, o.w);
            }
            rec4[(size_t)(rowbase + rl) * (DD / 4) + d4] = o;
        }
    }
}

extern "C" void kernel_launch(void* const* d_in, const int* in_sizes, int n_in,
                              void* d_out, int out_size, void* d_ws, size_t ws_size,
                              hipStream_t stream) {
    const float* x     = (const float*)d_in[0];
    // d_in[1] = is_protected (unused: reference applies alpha_p to all rows)
    const float* alpha = (const float*)d_in[2];
    const float* clw   = (const float*)d_in[3];
    const float* cent  = (const float*)d_in[4];

    float* out      = (float*)d_out;
    float* out_map  = out;                               // N*K
    float* out_rec  = out + (size_t)NN * KK;             // N*D (16B-aligned: N*K*4 = 1310720)
    float* out_pred = out_rec + (size_t)NN * DD;         // N

    dim3 grid(NN / ROWS_PB), block(256);
    lfr_fused_kernel<<<grid, block, 0, stream>>>(x, alpha, clw, cent,
                                                 out_map, out_rec, out_pred);
}